// MessageAggregation_81003083203131
// MI455X (gfx1250) — compile-verified
//
#include <hip/hip_runtime.h>

typedef __attribute__((ext_vector_type(2))) float v2f;
typedef __attribute__((ext_vector_type(4))) float v4f;
typedef __attribute__((ext_vector_type(8))) float v8f;

#define B_ROWS 8192
#define N_MATS 64
#define D_DIM  128
#define NEG_SLOPE 0.01f

// ---------------------------------------------------------------------------
// Kernel 1: bandwidth kernel.  s = sum_n e2[n];  a = e1 + s;  m = e1 * s
// Streams 268 MB of all_embeddings2 once with non-temporal b128 loads (zero
// reuse -> TH=NT keeps L2 free for a/m scratch + W, which ARE reused).
// One thread per 4 consecutive floats of the [B,D] plane.
// ---------------------------------------------------------------------------
__global__ void __launch_bounds__(256)
ma_reduce_prep(const float* __restrict__ e1,
               const float* __restrict__ e2,
               float* __restrict__ a,
               float* __restrict__ m) {
    const size_t idx = (size_t)blockIdx.x * blockDim.x + threadIdx.x; // v4f index
    const size_t plane4 = (size_t)B_ROWS * D_DIM / 4;                // 262144
    if (idx >= plane4) return;

    const v4f* __restrict__ e2v = (const v4f*)e2;

    v4f s = {0.f, 0.f, 0.f, 0.f};
#pragma unroll 8
    for (int n = 0; n < N_MATS; ++n) {
        v4f v = __builtin_nontemporal_load(&e2v[(size_t)n * plane4 + idx]);
        s += v;
    }

    v4f e = __builtin_nontemporal_load(&((const v4f*)e1)[idx]);

    // Regular-temporal stores: kernel 2 re-reads a/m from L2 immediately.
    ((v4f*)a)[idx] = e + s;
    ((v4f*)m)[idx] = e * s;
}

// ---------------------------------------------------------------------------
// Kernel 2: out = leakyrelu(a @ W1^T + m @ W2^T) using V_WMMA_F32_16X16X4_F32.
// One wave32 per 16x16 output tile; 32 k-steps of K=4, 2 WMMAs per step.
//
// Fragment layouts (ISA 7.12.2):
//   A 16x4:  lanes 0-15 -> row=lane, VGPR{0,1} = K{0,1}; lanes 16-31 -> K{2,3}
//   B 4x16:  lanes 0-15 -> col=lane, VGPR{0,1} = K{0,1}; lanes 16-31 -> K{2,3}
//   C 16x16: VGPR r -> row = r + 8*(lane/16), col = lane%16
// B-matrix element (k,n) = W[n*128 + k]  (W is [out,in] row-major) -> both A
// and B fragments are contiguous 8-byte loads.
// ---------------------------------------------------------------------------
__global__ void __launch_bounds__(128)
ma_gemm_lrelu(const float* __restrict__ a,
              const float* __restrict__ m,
              const float* __restrict__ W1,
              const float* __restrict__ W2,
              float* __restrict__ out) {
    const int wave    = threadIdx.x >> 5;
    const int lane    = threadIdx.x & 31;
    const int tile    = blockIdx.x * 4 + wave;     // 4096 tiles total, exact
    const int tiles_n = D_DIM / 16;                // 8
    const int tile_n  = tile % tiles_n;
    const int tile_m  = tile / tiles_n;

    const int half = lane >> 4;                    // 0: K{0,1}, 1: K{2,3}
    const int l    = lane & 15;
    const int arow = tile_m * 16 + l;              // A-matrix row for this lane
    const int bcol = tile_n * 16 + l;              // B-matrix col == W row (o)

    const float* __restrict__ aptr  = a  + (size_t)arow * D_DIM + 2 * half;
    const float* __restrict__ mptr  = m  + (size_t)arow * D_DIM + 2 * half;
    const float* __restrict__ w1ptr = W1 + (size_t)bcol * D_DIM + 2 * half;
    const float* __restrict__ w2ptr = W2 + (size_t)bcol * D_DIM + 2 * half;

    v8f c = {0.f, 0.f, 0.f, 0.f, 0.f, 0.f, 0.f, 0.f};

#pragma unroll 4
    for (int k0 = 0; k0 < D_DIM; k0 += 4) {
        v2f afrag = *(const v2f*)(aptr  + k0);
        v2f mfrag = *(const v2f*)(mptr  + k0);
        v2f w1f   = *(const v2f*)(w1ptr + k0);
        v2f w2f   = *(const v2f*)(w2ptr + k0);
        c = __builtin_amdgcn_wmma_f32_16x16x4_f32(
                false, afrag, false, w1f, (short)0, c, false, false);
        c = __builtin_amdgcn_wmma_f32_16x16x4_f32(
                false, mfrag, false, w2f, (short)0, c, false, false);
    }

    // LeakyReLU + store.  VGPR r holds (row = tile_m*16 + r + 8*half, col = bcol)
    float* __restrict__ optr = out + (size_t)(tile_m * 16 + 8 * half) * D_DIM
                                   + tile_n * 16 + l;
#pragma unroll
    for (int r = 0; r < 8; ++r) {
        float v = c[r];
        v = (v > 0.f) ? v : NEG_SLOPE * v;
        optr[(size_t)r * D_DIM] = v;
    }
}

// ---------------------------------------------------------------------------
extern "C" void kernel_launch(void* const* d_in, const int* in_sizes, int n_in,
                              void* d_out, int out_size, void* d_ws, size_t ws_size,
                              hipStream_t stream) {
    const float* e1 = (const float*)d_in[0];           // [B, D]
    const float* e2 = (const float*)d_in[1];           // [N, B, D]
    const float* W1 = (const float*)d_in[2];           // [D, D]
    const float* W2 = (const float*)d_in[3];           // [D, D]
    float* out = (float*)d_out;                        // [B, D]

    float* a = (float*)d_ws;                           // [B, D] scratch
    float* m = a + (size_t)B_ROWS * D_DIM;             // [B, D] scratch

    // Kernel 1: 262144 v4f elements / 256 threads = 1024 blocks
    ma_reduce_prep<<<(B_ROWS * D_DIM / 4 + 255) / 256, 256, 0, stream>>>(e1, e2, a, m);

    // Kernel 2: 4096 tiles, 4 waves (128 threads) per block = 1024 blocks
    ma_gemm_lrelu<<<1024, 128, 0, stream>>>(a, m, W1, W2, out);
}